// LiquidNeuronEncoder_71906342469697
// MI455X (gfx1250) — compile-verified
//
#include <hip/hip_runtime.h>
#include <hip/hip_bf16.h>
#include <math.h>

// ---------------------------------------------------------------------------
// LiquidNeuronEncoder on MI455X (gfx1250, wave32)
//
// reference math (only last timestep is live; wh_w is dead):
//   x[s]     = input_seq[s, T-1, 0]
//   delta[s] = input_seq[s, T-1, 1]
//   h_new[s,h] = delta[s] * tanh(x[s]*in_w[h] + in_b[h] + wh_b[h]) / tau[h]
//   out[s,l]   = tanh( sum_h h_new[s,h] * out_w[l,h] + out_b[l] )
//
// The GEMM (8192x64)·(64x64) runs on V_WMMA_F32_16X16X4_F32 (full fp32).
// One wave32 owns one 16-row x 64-col output tile.
// ---------------------------------------------------------------------------

typedef __attribute__((ext_vector_type(2))) float v2f;
typedef __attribute__((ext_vector_type(8))) float v8f;

#define S_DIM 8192
#define T_DIM 2048
#define D_DIM 2
#define H_DIM 64
#define L_DIM 64

#define WAVES_PER_BLOCK 8
#define TILE_ROWS 16
#define NUM_TILES (S_DIM / TILE_ROWS)   // 512

__global__ __launch_bounds__(32 * WAVES_PER_BLOCK)
void liquid_neuron_wmma_kernel(const float* __restrict__ input_seq,
                               const float* __restrict__ in_w,
                               const float* __restrict__ in_b,
                               const float* __restrict__ wh_b,
                               const float* __restrict__ tau,
                               const float* __restrict__ out_w,
                               const float* __restrict__ out_b,
                               float* __restrict__ out)
{
    const int lane = threadIdx.x & 31;
    const int wave = threadIdx.x >> 5;
    const int tile = blockIdx.x * WAVES_PER_BLOCK + wave;   // 16-row tile id
    if (tile >= NUM_TILES) return;

    const int row0 = tile * TILE_ROWS;
    const int m    = lane & 15;          // matrix row within tile (A layout)
    const int p    = (lane >> 4) << 1;   // K sub-pair: lanes 0-15 -> k0,k0+1 ; 16-31 -> k0+2,k0+3
    const int s    = row0 + m;

    // ---- load last timestep (x, delta) for this sample: 8B, 8B-aligned ----
    const float2 xd = *(const float2*)(input_seq +
                        (size_t)s * (T_DIM * D_DIM) + (size_t)(T_DIM - 1) * D_DIM);
    const float x     = xd.x;
    const float delta = xd.y;

    // ---- phase 1: liquid-neuron activation, produced directly in the
    //      V_WMMA_F32_16X16X4_F32 A-operand register layout.
    //      A is 16x4 per step: lane L holds row m=L&15, K-pair p..p+1. ----
    v2f a[16];
#pragma unroll
    for (int j = 0; j < 16; ++j) {
        const int h0 = 4 * j + p;
        const int h1 = h0 + 1;
        const float pre0 = fmaf(x, in_w[h0], in_b[h0] + wh_b[h0]);
        const float pre1 = fmaf(x, in_w[h1], in_b[h1] + wh_b[h1]);
        a[j].x = delta * (tanhf(pre0) / tau[h0]);
        a[j].y = delta * (tanhf(pre1) / tau[h1]);
    }

    // ---- phase 2: (16x64) @ (64x64)^T via 16 K-steps x 4 N-tiles of
    //      v_wmma_f32_16x16x4_f32. B[k][n] = out_w[n_global*H + k];
    //      lane n loads two consecutive k values (contiguous float2). ----
    const int n = lane & 15;
    v8f acc0 = {}, acc1 = {}, acc2 = {}, acc3 = {};
#pragma unroll
    for (int j = 0; j < 16; ++j) {
        const int k0 = 4 * j + p;
        const v2f b0 = *(const v2f*)(out_w + (size_t)(0  + n) * H_DIM + k0);
        const v2f b1 = *(const v2f*)(out_w + (size_t)(16 + n) * H_DIM + k0);
        const v2f b2 = *(const v2f*)(out_w + (size_t)(32 + n) * H_DIM + k0);
        const v2f b3 = *(const v2f*)(out_w + (size_t)(48 + n) * H_DIM + k0);
        // 8 args: (neg_a, A, neg_b, B, c_mod, C, reuse_a, reuse_b)
        acc0 = __builtin_amdgcn_wmma_f32_16x16x4_f32(false, a[j], false, b0,
                                                     (short)0, acc0, false, false);
        acc1 = __builtin_amdgcn_wmma_f32_16x16x4_f32(false, a[j], false, b1,
                                                     (short)0, acc1, false, false);
        acc2 = __builtin_amdgcn_wmma_f32_16x16x4_f32(false, a[j], false, b2,
                                                     (short)0, acc2, false, false);
        acc3 = __builtin_amdgcn_wmma_f32_16x16x4_f32(false, a[j], false, b3,
                                                     (short)0, acc3, false, false);
    }

    // ---- epilogue: C/D layout is VGPR r -> row (r + 8*(lane>>4)), col lane&15.
    //      Apply bias + tanh and store; each half-wave writes a contiguous
    //      64B segment per (r, n-tile). ----
    const int mo = (lane >> 4) << 3;     // 0 or 8
#pragma unroll
    for (int r = 0; r < 8; ++r) {
        float* o = out + (size_t)(row0 + mo + r) * L_DIM;
        o[n     ] = tanhf(acc0[r] + out_b[n     ]);
        o[n + 16] = tanhf(acc1[r] + out_b[n + 16]);
        o[n + 32] = tanhf(acc2[r] + out_b[n + 32]);
        o[n + 48] = tanhf(acc3[r] + out_b[n + 48]);
    }
}

extern "C" void kernel_launch(void* const* d_in, const int* in_sizes, int n_in,
                              void* d_out, int out_size, void* d_ws, size_t ws_size,
                              hipStream_t stream) {
    // setup_inputs() order:
    //   0 input_seq (S*T*D f32), 1 in_w (H), 2 in_b (H), 3 wh_w (H*H, dead),
    //   4 wh_b (H), 5 tau (H), 6 out_w (L*H), 7 out_b (L)
    const float* input_seq = (const float*)d_in[0];
    const float* in_w      = (const float*)d_in[1];
    const float* in_b      = (const float*)d_in[2];
    const float* wh_b      = (const float*)d_in[4];
    const float* tau       = (const float*)d_in[5];
    const float* out_w     = (const float*)d_in[6];
    const float* out_b     = (const float*)d_in[7];
    float* out             = (float*)d_out;

    (void)in_sizes; (void)n_in; (void)out_size; (void)d_ws; (void)ws_size;

    const int blocks = NUM_TILES / WAVES_PER_BLOCK;   // 512/8 = 64
    liquid_neuron_wmma_kernel<<<blocks, 32 * WAVES_PER_BLOCK, 0, stream>>>(
        input_seq, in_w, in_b, wh_b, tau, out_w, out_b, out);
}